// GraphNeuralNetwork_73435350827477
// MI455X (gfx1250) — compile-verified
//
#include <hip/hip_runtime.h>

#define N_NODES   50000
#define N_EDGES   800000
#define N_GRAPHS  64
#define IN_DIM    100
#define HID_DIM   128
#define OUT_DIM   256
#define EPSV      1e-5f

typedef __attribute__((ext_vector_type(2))) float v2f;
typedef __attribute__((ext_vector_type(8))) float v8f;

// ---------------------------------------------------------------- utilities
__global__ void fill_f32(float* __restrict__ p, float v, int n) {
  int i = blockIdx.x * blockDim.x + threadIdx.x;
  if (i < n) p[i] = v;
}

__global__ void count_nodes(const int* __restrict__ batch, float* __restrict__ gcnt, int n) {
  int i = blockIdx.x * blockDim.x + threadIdx.x;
  if (i < n) atomicAdd(&gcnt[batch[i]], 1.0f);
}

__global__ void deg_edges(const int* __restrict__ dst, float* __restrict__ deg, int e) {
  int i = blockIdx.x * blockDim.x + threadIdx.x;
  if (i < e) atomicAdd(&deg[dst[i]], 1.0f);
}

__global__ void rsqrt_k(const float* __restrict__ deg, float* __restrict__ dinv, int n) {
  int i = blockIdx.x * blockDim.x + threadIdx.x;
  if (i < n) dinv[i] = rsqrtf(deg[i]);
}

// ------------------------------------------------------------ WMMA f32 GEMM
// Y[N,F] = X[N,K] @ W[K,F].  One wave (32 lanes) computes one 16x16 tile,
// accumulating with V_WMMA_F32_16X16X4_F32 over K in steps of 4.
// Layout per CDNA5 ISA 7.12.2:
//   A 16x4 : lanes 0-15 -> M=lane, K=k+{0,1}; lanes 16-31 -> M=lane-16, K=k+{2,3}
//   B 4x16 : lanes 0-15 -> N=lane, K=k+{0,1}; lanes 16-31 -> N=lane-16, K=k+{2,3}
//   C/D    : VGPR v -> M = v + 8*(lane>=16), N = lane&15
__global__ void gemm_wmma_f32(const float* __restrict__ X, const float* __restrict__ W,
                              float* __restrict__ Y, int K, int F, int ldx) {
  const int lane = threadIdx.x;          // 0..31
  const int half = lane >> 4;            // 0 or 1
  const int lid  = lane & 15;
  const int m0   = blockIdx.x * 16;
  const int n0   = (blockIdx.y * 4 + threadIdx.y) * 16;

  const float* xrow = X + (size_t)(m0 + lid) * ldx + 2 * half;
  const float* wcol = W + (size_t)(2 * half) * F + n0 + lid;

  v8f acc = {};
  for (int k = 0; k < K; k += 4) {
    v2f a, b;
    a.x = xrow[k];
    a.y = xrow[k + 1];
    b.x = wcol[(size_t)k * F];
    b.y = wcol[(size_t)k * F + F];
    acc = __builtin_amdgcn_wmma_f32_16x16x4_f32(
        /*neg_a=*/false, a, /*neg_b=*/false, b,
        /*c_mod=*/(short)0, acc, /*reuse_a=*/false, /*reuse_b=*/false);
  }

  float* y = Y + ((size_t)m0 + 8 * half) * F + n0 + lid;
#pragma unroll
  for (int v = 0; v < 8; ++v) y[(size_t)v * F] = acc[v];
}

// -------------------------------------------------------------- GCN pieces
// out = xw * (1/deg) + bias  (self-loop term; scatter accumulates on top)
__global__ void selfloop_bias(const float* __restrict__ xw, const float* __restrict__ dinv,
                              const float* __restrict__ bias, float* __restrict__ out,
                              int F, int total) {
  int i = blockIdx.x * blockDim.x + threadIdx.x;
  if (i >= total) return;
  int n = i / F, f = i - n * F;
  float di = dinv[n];
  out[i] = xw[i] * di * di + bias[f];
}

// per-edge message scatter: out[dst] += xw[src] * dinv[src]*dinv[dst]
// thread t -> (edge, 4-feature chunk); consecutive t = same edge, coalesced.
__global__ void scatter_edges(const int* __restrict__ src, const int* __restrict__ dst,
                              const float* __restrict__ xw, const float* __restrict__ dinv,
                              float* __restrict__ out, int nedges, int F) {
  int t = blockIdx.x * blockDim.x + threadIdx.x;
  int fc = F >> 2;
  int e = t / fc;
  if (e >= nedges) return;
  int f = (t - e * fc) << 2;
  int s = src[e], d = dst[e];
  float c = dinv[s] * dinv[d];
  float4 v = *(const float4*)(xw + (size_t)s * F + f);
  float* o = out + (size_t)d * F + f;
  atomicAdd(o + 0, v.x * c);
  atomicAdd(o + 1, v.y * c);
  atomicAdd(o + 2, v.z * c);
  atomicAdd(o + 3, v.w * c);
}

// ------------------------------------------------------------ GraphNorm
__global__ void seg_sum(const float* __restrict__ x, const int* __restrict__ batch,
                        float* __restrict__ gsum, int F, int total) {
  int i = blockIdx.x * blockDim.x + threadIdx.x;
  if (i >= total) return;
  int n = i / F, f = i - n * F;
  atomicAdd(&gsum[batch[n] * F + f], x[i]);
}

__global__ void center_var(const float* __restrict__ x, const int* __restrict__ batch,
                           const float* __restrict__ gsum, const float* __restrict__ gcnt,
                           const float* __restrict__ ms, float* __restrict__ cent,
                           float* __restrict__ gvar, int F, int total) {
  int i = blockIdx.x * blockDim.x + threadIdx.x;
  if (i >= total) return;
  int n = i / F, f = i - n * F;
  int g = batch[n];
  float mean = gsum[g * F + f] / fmaxf(gcnt[g], 1.0f);
  float c = x[i] - mean * ms[f];
  cent[i] = c;
  atomicAdd(&gvar[g * F + f], c * c);
}

__global__ void norm_relu(const float* __restrict__ cent, const int* __restrict__ batch,
                          const float* __restrict__ gvar, const float* __restrict__ gcnt,
                          const float* __restrict__ w, const float* __restrict__ b,
                          float* __restrict__ act, int F, int total) {
  int i = blockIdx.x * blockDim.x + threadIdx.x;
  if (i >= total) return;
  int n = i / F, f = i - n * F;
  int g = batch[n];
  float var = gvar[g * F + f] / fmaxf(gcnt[g], 1.0f);
  float o = w[f] * cent[i] * rsqrtf(var + EPSV) + b[f];
  act[i] = fmaxf(o, 0.0f);
}

// ------------------------------------------------------------ pooling
// post-ReLU values are >= 0, so uint bit-pattern ordering == float ordering,
// and init 0 reproduces the reference's (-inf -> 0) empty-graph behavior.
__global__ void pool_acc(const float* __restrict__ h, const int* __restrict__ batch,
                         float* __restrict__ gsum, unsigned* __restrict__ gmax,
                         int F, int total) {
  int i = blockIdx.x * blockDim.x + threadIdx.x;
  if (i >= total) return;
  int n = i / F, f = i - n * F;
  int g = batch[n];
  float v = h[i];
  atomicAdd(&gsum[g * F + f], v);
  atomicMax(&gmax[g * F + f], __float_as_uint(v));
}

__global__ void pool_out(const float* __restrict__ gsum, const unsigned* __restrict__ gmax,
                         const float* __restrict__ gcnt, float* __restrict__ out, int total) {
  int i = blockIdx.x * blockDim.x + threadIdx.x;
  if (i >= total) return;
  int g = i / OUT_DIM, f = i - g * OUT_DIM;
  out[(size_t)g * (2 * OUT_DIM) + f] = gsum[i] / fmaxf(gcnt[g], 1.0f);
  out[(size_t)g * (2 * OUT_DIM) + OUT_DIM + f] = __uint_as_float(gmax[i]);
}

// ------------------------------------------------------------ launcher
extern "C" void kernel_launch(void* const* d_in, const int* in_sizes, int n_in,
                              void* d_out, int out_size, void* d_ws, size_t ws_size,
                              hipStream_t stream) {
  const float* x     = (const float*)d_in[0];
  const int*   ei    = (const int*)d_in[1];
  const int*   src   = ei;
  const int*   dst   = ei + N_EDGES;
  const int*   batch = (const int*)d_in[2];
  const float* Ws[3] = {(const float*)d_in[3], (const float*)d_in[5], (const float*)d_in[7]};
  const float* bs[3] = {(const float*)d_in[4], (const float*)d_in[6], (const float*)d_in[8]};
  const float* gw[3]  = {(const float*)d_in[9],  (const float*)d_in[12], (const float*)d_in[15]};
  const float* gb[3]  = {(const float*)d_in[10], (const float*)d_in[13], (const float*)d_in[16]};
  const float* gms[3] = {(const float*)d_in[11], (const float*)d_in[14], (const float*)d_in[17]};

  const size_t NF = (size_t)N_NODES * OUT_DIM;  // max per-layer activation size
  float*    bufA = (float*)d_ws;                // layer input / output activation
  float*    bufB = bufA + NF;                   // xw, then centered values
  float*    bufC = bufB + NF;                   // conv output
  float*    deg  = bufC + NF;
  float*    dinv = deg + N_NODES;
  float*    gsum = dinv + N_NODES;
  float*    gvar = gsum + (size_t)N_GRAPHS * OUT_DIM;
  unsigned* gmax = (unsigned*)(gvar + (size_t)N_GRAPHS * OUT_DIM);
  float*    gcnt = (float*)(gmax + (size_t)N_GRAPHS * OUT_DIM);

  auto cdiv = [](int a, int b) { return (a + b - 1) / b; };

  // degrees (with self-loop +1) and per-graph node counts
  fill_f32<<<1, 64, 0, stream>>>(gcnt, 0.0f, N_GRAPHS);
  fill_f32<<<cdiv(N_NODES, 256), 256, 0, stream>>>(deg, 1.0f, N_NODES);
  count_nodes<<<cdiv(N_NODES, 256), 256, 0, stream>>>(batch, gcnt, N_NODES);
  deg_edges<<<cdiv(N_EDGES, 256), 256, 0, stream>>>(dst, deg, N_EDGES);
  rsqrt_k<<<cdiv(N_NODES, 256), 256, 0, stream>>>(deg, dinv, N_NODES);

  const float* cur = x;
  int K = IN_DIM, ld = IN_DIM;
  for (int l = 0; l < 3; ++l) {
    const int F   = (l == 2) ? OUT_DIM : HID_DIM;
    const int NFl = N_NODES * F;

    // 1) xw = act @ W   (WMMA f32 16x16x4)
    dim3 gg(N_NODES / 16, F / 64), gbk(32, 4);
    gemm_wmma_f32<<<gg, gbk, 0, stream>>>(cur, Ws[l], bufB, K, F, ld);

    // 2) conv = xw/deg + bias, then scatter edge messages
    selfloop_bias<<<cdiv(NFl, 256), 256, 0, stream>>>(bufB, dinv, bs[l], bufC, F, NFl);
    scatter_edges<<<cdiv(N_EDGES * (F / 4), 256), 256, 0, stream>>>(src, dst, bufB, dinv,
                                                                    bufC, N_EDGES, F);

    // 3) GraphNorm + ReLU
    fill_f32<<<cdiv(N_GRAPHS * F, 256), 256, 0, stream>>>(gsum, 0.0f, N_GRAPHS * F);
    fill_f32<<<cdiv(N_GRAPHS * F, 256), 256, 0, stream>>>(gvar, 0.0f, N_GRAPHS * F);
    seg_sum<<<cdiv(NFl, 256), 256, 0, stream>>>(bufC, batch, gsum, F, NFl);
    center_var<<<cdiv(NFl, 256), 256, 0, stream>>>(bufC, batch, gsum, gcnt, gms[l],
                                                   bufB, gvar, F, NFl);
    norm_relu<<<cdiv(NFl, 256), 256, 0, stream>>>(bufB, batch, gvar, gcnt, gw[l], gb[l],
                                                  bufA, F, NFl);
    cur = bufA;
    K = F;
    ld = F;
  }

  // mean + max pooling -> [64, 512]
  fill_f32<<<cdiv(N_GRAPHS * OUT_DIM, 256), 256, 0, stream>>>(gsum, 0.0f, N_GRAPHS * OUT_DIM);
  fill_f32<<<cdiv(N_GRAPHS * OUT_DIM, 256), 256, 0, stream>>>((float*)gmax, 0.0f,
                                                              N_GRAPHS * OUT_DIM);
  pool_acc<<<cdiv(N_NODES * OUT_DIM, 256), 256, 0, stream>>>(bufA, batch, gsum, gmax,
                                                             OUT_DIM, N_NODES * OUT_DIM);
  pool_out<<<cdiv(N_GRAPHS * OUT_DIM, 256), 256, 0, stream>>>(gsum, gmax, gcnt,
                                                              (float*)d_out,
                                                              N_GRAPHS * OUT_DIM);
}